// PointPatchifyIdx_72241349919077
// MI455X (gfx1250) — compile-verified
//
#include <hip/hip_runtime.h>

#define BATCH   32
#define NPTS    8192
#define KCENT   128
#define PATCH   32
#define CHUNK   256
#define NCHUNK  (NPTS / CHUNK)
#define PITCH   257   // odd pitch: lane-half (m vs m+8) writes land in different banks

typedef __attribute__((ext_vector_type(2))) float v2f;
typedef __attribute__((ext_vector_type(8))) float v8f;

// ---------------------------------------------------------------------------
// Kernel 1: farthest point sampling. One 1024-thread workgroup per batch.
// Points + running min distances live entirely in registers (8 per thread).
// Emits centers[b][k][3] (coords of selected points) into d_out's tail.
// ---------------------------------------------------------------------------
__global__ __launch_bounds__(1024) void fps_kernel(const float* __restrict__ pts,
                                                   float* __restrict__ centers_out) {
  const int b = blockIdx.x;
  const int t = threadIdx.x;
  const int lane = t & 31, wid = t >> 5;
  const float* p = pts + (size_t)b * NPTS * 3;

  float px[8], py[8], pz[8], mind[8];
#pragma unroll
  for (int i = 0; i < 8; ++i) {
    int n = i * 1024 + t;
    px[i] = p[n * 3 + 0];
    py[i] = p[n * 3 + 1];
    pz[i] = p[n * 3 + 2];
    mind[i] = __builtin_inff();
  }

  __shared__ float s_red_v[32];
  __shared__ int   s_red_i[32];
  __shared__ float s_b[3];
  __shared__ int   s_nxt;

  if (t == 0) { s_b[0] = px[0]; s_b[1] = py[0]; s_b[2] = pz[0]; }  // start idx 0

  float* cout = centers_out + (size_t)b * KCENT * 3;

  for (int k = 0; k < KCENT; ++k) {
    __syncthreads();                       // makes s_b (last center) visible
    float lx = s_b[0], ly = s_b[1], lz = s_b[2];
    if (t == 0) { cout[k * 3 + 0] = lx; cout[k * 3 + 1] = ly; cout[k * 3 + 2] = lz; }

    // update running min + local argmax over this thread's 8 points
    float bv = -1.0f; int bi = 0x7fffffff;
#pragma unroll
    for (int i = 0; i < 8; ++i) {
      float dx = px[i] - lx, dy = py[i] - ly, dz = pz[i] - lz;
      float d  = fmaf(dx, dx, fmaf(dy, dy, dz * dz));
      mind[i]  = fminf(mind[i], d);
      int n = i * 1024 + t;
      if (mind[i] > bv || (mind[i] == bv && n < bi)) { bv = mind[i]; bi = n; }
    }
    // wave32 argmax reduction
#pragma unroll
    for (int off = 16; off > 0; off >>= 1) {
      float ov = __shfl_down(bv, off, 32);
      int   oi = __shfl_down(bi, off, 32);
      if (ov > bv || (ov == bv && oi < bi)) { bv = ov; bi = oi; }
    }
    if (lane == 0) { s_red_v[wid] = bv; s_red_i[wid] = bi; }
    __syncthreads();
    if (wid == 0) {
      bv = s_red_v[lane]; bi = s_red_i[lane];
#pragma unroll
      for (int off = 16; off > 0; off >>= 1) {
        float ov = __shfl_down(bv, off, 32);
        int   oi = __shfl_down(bi, off, 32);
        if (ov > bv || (ov == bv && oi < bi)) { bv = ov; bi = oi; }
      }
      if (lane == 0) s_nxt = bi;
    }
    __syncthreads();
    int nxt = s_nxt;
    if ((nxt & 1023) == t) {               // owner broadcasts new center coords
      int i = nxt >> 10;
      s_b[0] = px[i]; s_b[1] = py[i]; s_b[2] = pz[i];
    }
  }
}

// ---------------------------------------------------------------------------
// Kernel 2: WMMA distance tiles + wave-cooperative top-32 selection.
// Grid: (KCENT/16, BATCH); 512 threads = 16 waves. Wave w computes one
// 16x16 WMMA tile per 256-column chunk and owns the selection for center w.
// Selection key s = |p|^2 - 2 c.p  (|c|^2 is a per-row constant -> same order).
// Software-pipelined point loads + double-buffered LDS tile (1 barrier/chunk).
// ---------------------------------------------------------------------------
__global__ __launch_bounds__(512) void knn_kernel(const float* __restrict__ pts,
                                                  const float* __restrict__ centers,
                                                  int* __restrict__ idx_out) {
  const int tile = blockIdx.x;          // 16-center tile
  const int b    = blockIdx.y;
  const int t    = threadIdx.x;
  const int lane = t & 31, w = t >> 5;  // 16 waves
  const int half = lane >> 4, n16 = lane & 15;

  __shared__ float s_d[2][16 * PITCH];

  const float* p = pts + (size_t)b * NPTS * 3;
  const float* c = centers + ((size_t)b * KCENT + (size_t)tile * 16) * 3;

  // A fragment: 16x4 matrix [cx cy cz 1]; lanes 0-15 hold K0,K1; 16-31 hold K2,K3
  float cx = c[n16 * 3 + 0], cy = c[n16 * 3 + 1], cz = c[n16 * 3 + 2];
  v2f afrag;
  afrag[0] = half ? cz : cx;
  afrag[1] = half ? 1.0f : cy;

  // per-lane slot of this wave's sorted best-32 list (ascending; lane31 = max)
  float list_d = __builtin_inff();
  int   list_i = 0;
  float curmax = __builtin_inff();

  // prologue: load this wave's point column for chunk 0
  int n0 = w * 16 + n16;
  float qx = p[n0 * 3 + 0], qy = p[n0 * 3 + 1], qz = p[n0 * 3 + 2];

  for (int cix = 0; cix < NCHUNK; ++cix) {
    const int base = cix * CHUNK;

    // ---- prefetch next chunk's point column (consumed next iteration) ----
    float nx = 0.0f, ny = 0.0f, nz = 0.0f;
    if (cix + 1 < NCHUNK) {               // uniform branch: EXEC stays all-1s
      int nn = base + CHUNK + w * 16 + n16;
      nx = p[nn * 3 + 0]; ny = p[nn * 3 + 1]; nz = p[nn * 3 + 2];
    }

    // ---- compute: wave w fills columns [w*16, w*16+16) of the chunk ----
    float ss = fmaf(qx, qx, fmaf(qy, qy, qz * qz));
    v2f bfrag;                            // B: 4x16 matrix [-2px -2py -2pz |p|^2]
    bfrag[0] = half ? (-2.0f * qz) : (-2.0f * qx);
    bfrag[1] = half ? ss            : (-2.0f * qy);
    v8f acc = {};
    acc = __builtin_amdgcn_wmma_f32_16x16x4_f32(false, afrag, false, bfrag,
                                                (short)0, acc, false, false);
    float* buf = s_d[cix & 1];
#pragma unroll
    for (int i = 0; i < 8; ++i) {         // D layout: row = half*8+i, col = lane&15
      int mm = half * 8 + i;
      buf[mm * PITCH + w * 16 + n16] = acc[i];
    }
    __syncthreads();   // buf[cix&1] complete; also guards buf reuse (see proof)

    // ---- select: wave w scans its row; batch the 8 DS loads first ----
    float dv[8];
#pragma unroll
    for (int r = 0; r < 8; ++r) dv[r] = buf[w * PITCH + r * 32 + lane];
#pragma unroll
    for (int r = 0; r < 8; ++r) {
      float d  = dv[r];
      int   gi = base + r * 32 + lane;
      unsigned mask = __builtin_amdgcn_ballot_w32(d < curmax);
      while (mask) {                      // wave-uniform loop over qualifying lanes
        int j = __builtin_ctz(mask);
        mask &= mask - 1;
        float dj = __shfl(d, j, 32);
        int   ij = __shfl(gi, j, 32);
        if (dj < curmax) {                // recheck: curmax shrinks as we insert
          unsigned lt = __builtin_amdgcn_ballot_w32(list_d < dj);
          int pos = __builtin_popcount(lt);
          float pd = __shfl_up(list_d, 1, 32);
          int   pi = __shfl_up(list_i, 1, 32);
          if (lane >= pos) {
            list_d = (lane == pos) ? dj : pd;
            list_i = (lane == pos) ? ij : pi;
          }
          curmax = __shfl(list_d, 31, 32);
        }
      }
    }

    qx = nx; qy = ny; qz = nz;            // rotate pipeline
  }

  // lane = rank (ascending distance), matching top_k's sorted output
  idx_out[((size_t)b * KCENT + (size_t)tile * 16 + w) * PATCH + lane] = list_i;
}

// ---------------------------------------------------------------------------
extern "C" void kernel_launch(void* const* d_in, const int* in_sizes, int n_in,
                              void* d_out, int out_size, void* d_ws, size_t ws_size,
                              hipStream_t stream) {
  const float* pts = (const float*)d_in[0];
  int*   idx_out     = (int*)d_out;                                    // [32,128,32] i32
  float* centers_out = (float*)d_out + (size_t)BATCH * KCENT * PATCH;  // [32,128,3] f32

  fps_kernel<<<dim3(BATCH), dim3(1024), 0, stream>>>(pts, centers_out);
  knn_kernel<<<dim3(KCENT / 16, BATCH), dim3(512), 0, stream>>>(pts, centers_out, idx_out);
}